// NeuralODE_20761871909679
// MI455X (gfx1250) — compile-verified
//
#include <hip/hip_runtime.h>
#include <math.h>

typedef __attribute__((ext_vector_type(2))) float v2f;
typedef __attribute__((ext_vector_type(8))) float v8f;

#define B_ 8192
#define D_ 128
#define N_ELEM (B_ * D_)
#define N_MAX_STEPS 48
#define BROWS 64      // rows per block
#define LPAD 132      // padded LDS row stride (floats) to avoid bank conflicts

// Replay the clamped time accumulation exactly as the reference scan does:
// tt_{i+1} = tt_i + clip(T - tt_i, 0, dt0); returns dt for `step`.
__device__ __forceinline__ float step_dt(float dt0, float T, int step) {
    float tt = 0.0f;
    for (int i = 0; i < step; ++i)
        tt += fminf(dt0, fmaxf(0.0f, T - tt));
    return fminf(dt0, fmaxf(0.0f, T - tt));
}

// ---------------------------------------------------------------------------
// Fused MLP eval: Out = tanh((Y + dt*(c1*K1+..+c5*K5)) @ W1 + b1) @ W2 + b2
// Grid: 128 blocks x 128 threads (4 waves, 16 rows per wave).
// ---------------------------------------------------------------------------
__global__ __launch_bounds__(128) void mlp_eval(
    const float* __restrict__ Y,
    const float* __restrict__ K1, const float* __restrict__ K2,
    const float* __restrict__ K3, const float* __restrict__ K4,
    const float* __restrict__ K5,
    float c1, float c2, float c3, float c4, float c5,
    const float* __restrict__ W1, const float* __restrict__ b1,
    const float* __restrict__ W2, const float* __restrict__ b2,
    float* __restrict__ Out,
    const float* __restrict__ dt0p, const float* __restrict__ tp,
    int step)
{
    __shared__ float sY[BROWS * LPAD];

    const float dt0 = dt0p[0];
    const float T   = tp[0] / 10.0f;   // TIMESCALE
    const float dt  = step_dt(dt0, T, step);
    if (dt <= 0.0f) return;            // uniform: whole step is identity

    const int tid  = threadIdx.x;
    const int rowg = blockIdx.x * BROWS;

    // Stage combined input tile into LDS (coalesced global reads).
    for (int i = tid; i < BROWS * D_; i += 128) {
        const int r = i >> 7, c = i & 127;
        const int g = (rowg + r) * D_ + c;
        float a = 0.0f;
        if (c1 != 0.0f) a += c1 * K1[g];
        if (c2 != 0.0f) a += c2 * K2[g];
        if (c3 != 0.0f) a += c3 * K3[g];
        if (c4 != 0.0f) a += c4 * K4[g];
        if (c5 != 0.0f) a += c5 * K5[g];
        sY[r * LPAD + c] = Y[g] + dt * a;
    }
    __syncthreads();

    const int lane = tid & 31;
    const int wid  = tid >> 5;
    const int m0   = lane & 15;       // row within wave tile (A), col (B/C/D)
    const int hi   = lane >> 4;       // half-wave select
    const int ko   = hi * 2;          // K offset for A/B fragments
    const int wrow = wid * 16;        // this wave's row stripe in the block

    v8f zero = {};
    v8f acc[8];

    // ---- GEMM1: hidden = tanh(Yin @ W1 + b1), hidden -> own LDS rows ----
    #pragma unroll
    for (int nt = 0; nt < 8; ++nt) acc[nt] = zero;

    for (int kc = 0; kc < 128; kc += 4) {
        v2f a = *(const v2f*)&sY[(wrow + m0) * LPAD + kc + ko];
        #pragma unroll
        for (int nt = 0; nt < 8; ++nt) {
            v2f b;
            b.x = W1[(kc + ko)     * D_ + nt * 16 + m0];
            b.y = W1[(kc + ko + 1) * D_ + nt * 16 + m0];
            acc[nt] = __builtin_amdgcn_wmma_f32_16x16x4_f32(
                false, a, false, b, (short)0, acc[nt], false, false);
        }
    }
    #pragma unroll
    for (int nt = 0; nt < 8; ++nt) {
        const int n = nt * 16 + m0;
        const float bias = b1[n];
        #pragma unroll
        for (int r = 0; r < 8; ++r) {
            // C layout: VGPR r holds rows {r, r+8}; col = lane&15
            sY[(wrow + r + hi * 8) * LPAD + n] = tanhf(acc[nt][r] + bias);
        }
    }
    // Each wave reads/writes only its own 16 LDS rows -> no barrier needed.

    // ---- GEMM2: out = hidden @ W2 + b2 ----
    #pragma unroll
    for (int nt = 0; nt < 8; ++nt) acc[nt] = zero;

    for (int kc = 0; kc < 128; kc += 4) {
        v2f a = *(const v2f*)&sY[(wrow + m0) * LPAD + kc + ko];
        #pragma unroll
        for (int nt = 0; nt < 8; ++nt) {
            v2f b;
            b.x = W2[(kc + ko)     * D_ + nt * 16 + m0];
            b.y = W2[(kc + ko + 1) * D_ + nt * 16 + m0];
            acc[nt] = __builtin_amdgcn_wmma_f32_16x16x4_f32(
                false, a, false, b, (short)0, acc[nt], false, false);
        }
    }
    #pragma unroll
    for (int nt = 0; nt < 8; ++nt) {
        const int n = nt * 16 + m0;
        const float bias = b2[n];
        #pragma unroll
        for (int r = 0; r < 8; ++r) {
            Out[(rowg + wrow + r + hi * 8) * D_ + n] = acc[nt][r] + bias;
        }
    }
}

// ---------------------------------------------------------------------------
// Dopri5 combine: y += dt*(35/384 k1 + 500/1113 k3 + 125/192 k4
//                          - 2187/6784 k5 + 11/84 k6)
// ---------------------------------------------------------------------------
__global__ __launch_bounds__(256) void update_y(
    float* __restrict__ y,
    const float* __restrict__ k1, const float* __restrict__ k3,
    const float* __restrict__ k4, const float* __restrict__ k5,
    const float* __restrict__ k6,
    const float* __restrict__ dt0p, const float* __restrict__ tp, int step)
{
    const float dt0 = dt0p[0];
    const float T   = tp[0] / 10.0f;
    const float dt  = step_dt(dt0, T, step);
    if (dt <= 0.0f) return;
    const int i = blockIdx.x * blockDim.x + threadIdx.x;
    if (i < N_ELEM) {
        const float a = (float)(35.0 / 384.0)    * k1[i]
                      + (float)(500.0 / 1113.0)  * k3[i]
                      + (float)(125.0 / 192.0)   * k4[i]
                      - (float)(2187.0 / 6784.0) * k5[i]
                      + (float)(11.0 / 84.0)     * k6[i];
        y[i] += dt * a;
    }
}

// out[0:N] = x (SaveAt ts[0]); out[N:2N] = x = initial y (updated in place)
__global__ __launch_bounds__(256) void init_out(
    const float* __restrict__ x, float* __restrict__ out)
{
    const int i = blockIdx.x * blockDim.x + threadIdx.x;
    if (i < N_ELEM) {
        const float v = x[i];
        out[i] = v;
        out[N_ELEM + i] = v;
    }
}

// ---------------------------------------------------------------------------
// Initial step size heuristic on row 0 (single 128-thread block).
// ---------------------------------------------------------------------------
__device__ void row_mlp(const float* __restrict__ W1, const float* __restrict__ b1,
                        const float* __restrict__ W2, const float* __restrict__ b2,
                        const float* yin, float* hbuf, float* fout, int j)
{
    float s = b1[j];
    for (int k = 0; k < 128; ++k) s += yin[k] * W1[k * 128 + j];
    __syncthreads();                 // protect hbuf reuse
    hbuf[j] = tanhf(s);
    __syncthreads();
    float o = b2[j];
    for (int k = 0; k < 128; ++k) o += hbuf[k] * W2[k * 128 + j];
    fout[j] = o;
    __syncthreads();
}

__device__ float block_norm(float* red, float v, int j)
{
    __syncthreads();                 // protect red reuse
    red[j] = v * v;
    __syncthreads();
    for (int off = 64; off > 0; off >>= 1) {
        if (j < off) red[j] += red[j + off];
        __syncthreads();
    }
    return sqrtf(red[0]);
}

__global__ __launch_bounds__(128) void init_dt0(
    const float* __restrict__ tp, const float* __restrict__ x,
    const float* __restrict__ W1, const float* __restrict__ b1,
    const float* __restrict__ W2, const float* __restrict__ b2,
    float* __restrict__ dt0_out)
{
    (void)tp;
    __shared__ float y0[128], f0[128], y1[128], f1[128], hbuf[128], red[128];
    const int j = threadIdx.x;
    y0[j] = x[j];                    // row 0 of x
    __syncthreads();

    row_mlp(W1, b1, W2, b2, y0, hbuf, f0, j);

    const float sc = 1.4e-8f + fabsf(y0[j]) * 1.4e-8f;
    const float d0 = block_norm(red, y0[j] / sc, j);
    const float d1 = block_norm(red, f0[j] / sc, j);
    const float h0 = ((d0 < 1e-5f) || (d1 < 1e-5f)) ? 1e-6f : 0.01f * d0 / d1;

    __syncthreads();
    y1[j] = y0[j] + h0 * f0[j];
    __syncthreads();

    row_mlp(W1, b1, W2, b2, y1, hbuf, f1, j);

    const float d2 = block_norm(red, (f1[j] - f0[j]) / sc, j) / h0;
    float h1;
    if ((d1 <= 1e-15f) && (d2 <= 1e-15f)) h1 = fmaxf(1e-6f, h0 * 1e-3f);
    else                                  h1 = powf(0.01f / (d1 + d2), 0.2f);

    if (j == 0) dt0_out[0] = fminf(100.0f * h0, h1);
}

// ---------------------------------------------------------------------------
extern "C" void kernel_launch(void* const* d_in, const int* in_sizes, int n_in,
                              void* d_out, int out_size, void* d_ws, size_t ws_size,
                              hipStream_t stream)
{
    (void)in_sizes; (void)n_in; (void)out_size; (void)ws_size;
    const float* t  = (const float*)d_in[0];
    const float* x  = (const float*)d_in[1];
    const float* W1 = (const float*)d_in[2];
    const float* b1 = (const float*)d_in[3];
    const float* W2 = (const float*)d_in[4];
    const float* b2 = (const float*)d_in[5];
    float* out = (float*)d_out;
    float* ws  = (float*)d_ws;

    float* dt0 = ws;                       // 1 scalar (16-float aligned slot)
    float* k1 = ws + 16 + 0 * (size_t)N_ELEM;
    float* k2 = ws + 16 + 1 * (size_t)N_ELEM;
    float* k3 = ws + 16 + 2 * (size_t)N_ELEM;
    float* k4 = ws + 16 + 3 * (size_t)N_ELEM;
    float* k5 = ws + 16 + 4 * (size_t)N_ELEM;
    float* k6 = ws + 16 + 5 * (size_t)N_ELEM;
    float* y  = out + N_ELEM;              // y_T accumulates in-place in d_out

    const dim3 eg(N_ELEM / 256), eb(256);
    init_out<<<eg, eb, 0, stream>>>(x, out);
    init_dt0<<<dim3(1), dim3(128), 0, stream>>>(t, x, W1, b1, W2, b2, dt0);

    const dim3 mg(B_ / BROWS), mb(128);    // 128 blocks x 4 waves
    for (int s = 0; s < N_MAX_STEPS; ++s) {
        // k1 = f(y)
        mlp_eval<<<mg, mb, 0, stream>>>(y, y, y, y, y, y,
            0.0f, 0.0f, 0.0f, 0.0f, 0.0f,
            W1, b1, W2, b2, k1, dt0, t, s);
        // k2 = f(y + dt*(1/5 k1))
        mlp_eval<<<mg, mb, 0, stream>>>(y, k1, y, y, y, y,
            (float)(1.0 / 5.0), 0.0f, 0.0f, 0.0f, 0.0f,
            W1, b1, W2, b2, k2, dt0, t, s);
        // k3 = f(y + dt*(3/40 k1 + 9/40 k2))
        mlp_eval<<<mg, mb, 0, stream>>>(y, k1, k2, y, y, y,
            (float)(3.0 / 40.0), (float)(9.0 / 40.0), 0.0f, 0.0f, 0.0f,
            W1, b1, W2, b2, k3, dt0, t, s);
        // k4
        mlp_eval<<<mg, mb, 0, stream>>>(y, k1, k2, k3, y, y,
            (float)(44.0 / 45.0), (float)(-56.0 / 15.0), (float)(32.0 / 9.0),
            0.0f, 0.0f,
            W1, b1, W2, b2, k4, dt0, t, s);
        // k5
        mlp_eval<<<mg, mb, 0, stream>>>(y, k1, k2, k3, k4, y,
            (float)(19372.0 / 6561.0), (float)(-25360.0 / 2187.0),
            (float)(64448.0 / 6561.0), (float)(-212.0 / 729.0), 0.0f,
            W1, b1, W2, b2, k5, dt0, t, s);
        // k6
        mlp_eval<<<mg, mb, 0, stream>>>(y, k1, k2, k3, k4, k5,
            (float)(9017.0 / 3168.0), (float)(-355.0 / 33.0),
            (float)(46732.0 / 5247.0), (float)(49.0 / 176.0),
            (float)(-5103.0 / 18656.0),
            W1, b1, W2, b2, k6, dt0, t, s);
        // y += dt * (b1 k1 + b3 k3 + b4 k4 + b5 k5 + b6 k6)
        update_y<<<eg, eb, 0, stream>>>(y, k1, k3, k4, k5, k6, dt0, t, s);
    }
}